// SpatialTransformerPooled3d_14276471291940
// MI455X (gfx1250) — compile-verified
//
#include <hip/hip_runtime.h>
#include <hip/hip_bf16.h>

#if defined(__HIP_DEVICE_COMPILE__) && !__has_builtin(__builtin_amdgcn_wmma_f32_16x16x4_f32)
#error "missing __builtin_amdgcn_wmma_f32_16x16x4_f32 on this toolchain (device pass)"
#endif

#if defined(__HIP_DEVICE_COMPILE__) && \
    __has_builtin(__builtin_amdgcn_global_load_async_to_lds_b128) && \
    __has_builtin(__builtin_amdgcn_s_wait_asynccnt)
#define HAVE_ASYNC_LDS 1
#else
#define HAVE_ASYNC_LDS 0
#endif

typedef float v2f __attribute__((ext_vector_type(2)));
typedef float v8f __attribute__((ext_vector_type(8)));
// Matches the builtin's expected pointee type: int __attribute__((vector_size(16)))
typedef int v4i_vs __attribute__((vector_size(16)));
#define GPTR(p) ((__attribute__((address_space(1))) v4i_vs*)(p))
#define LPTR(p) ((__attribute__((address_space(3))) v4i_vs*)(p))

// Problem constants (match reference)
#define NB   16
#define TT   16
#define CC   64
#define WH   64
#define PP   2048
#define BTOT 256   // NB*TT batch-time rows

// Workspace layout (requires ws_size >= ~84 MB)
#define Z1_ELEMS (256u*64u*32u*32u)          // 16,777,216 floats
#define Z2_ELEMS (256u*64u*16u*16u)          //  4,194,304 floats
#define Z1_OFF   ((size_t)0)
#define Z2_OFF   ((size_t)Z1_ELEMS * 4)
#define META_OFF (Z2_OFF + (size_t)Z2_ELEMS * 4)

// ---------------- meta init: per level {xmin,xmax,ymin,ymax} ----------------
__global__ void k_init_meta(int* meta) {
    int i = threadIdx.x;
    if (i < 12) meta[i] = ((i & 1) == 0) ? 0x7FFFFFFF : (int)0x80000000;
}

// ---------------- bounding rect of all sample corners, per level ------------
__global__ void k_bbox(const float* __restrict__ grid, int* __restrict__ meta) {
    int p = blockIdx.x * blockDim.x + threadIdx.x;
    if (p >= PP) return;
    float gx = fminf(fmaxf(grid[2 * p + 0], -1.f), 1.f);
    float gy = fminf(fmaxf(grid[2 * p + 1], -1.f), 1.f);
#pragma unroll
    for (int l = 0; l < 3; ++l) {
        int W = WH >> l;
        float ix = ((gx + 1.f) * (float)W - 1.f) * 0.5f;
        float iy = ((gy + 1.f) * (float)W - 1.f) * 0.5f;
        int ix0 = (int)floorf(ix), iy0 = (int)floorf(iy);
        int xc0 = min(max(ix0, 0), W - 1), xc1 = min(max(ix0 + 1, 0), W - 1);
        int yc0 = min(max(iy0, 0), W - 1), yc1 = min(max(iy0 + 1, 0), W - 1);
        atomicMin(&meta[l * 4 + 0], xc0);
        atomicMax(&meta[l * 4 + 1], xc1);
        atomicMin(&meta[l * 4 + 2], yc0);
        atomicMax(&meta[l * 4 + 3], yc1);
    }
}

// ---------------- pyramid level 1: (256,64,32,32) from x --------------------
__global__ void k_pool1(const float* __restrict__ x, float* __restrict__ z1) {
    unsigned o = blockIdx.x * 256u + threadIdx.x;       // < 16,777,216
    unsigned xx = o & 31u, yy = (o >> 5) & 31u, c = (o >> 10) & 63u, bt = o >> 16;
    unsigned b = bt >> 4, t = bt & 15u;
    const float* src = x + (((b * 64u + c) * 16u + t) * 4096u) + (yy * 2u) * 64u + xx * 2u;
    z1[o] = 0.25f * (src[0] + src[1] + src[64] + src[65]);
}

// ---------------- pyramid level 2: (256,64,16,16) from z1 -------------------
__global__ void k_pool2(const float* __restrict__ z1, float* __restrict__ z2) {
    unsigned o = blockIdx.x * 256u + threadIdx.x;       // < 4,194,304
    unsigned xx = o & 15u, yy = (o >> 4) & 15u, c = (o >> 8) & 63u, bt = o >> 14;
    const float* src = z1 + (bt * 64u + c) * 1024u + (yy * 2u) * 32u + xx * 2u;
    z2[o] = 0.25f * (src[0] + src[1] + src[32] + src[33]);
}

// ---------------- main WMMA sampling kernel ---------------------------------
// Grid: (16 point-groups, 16 batch-tiles), block 256 = 8 wave32.
// Each wave computes a 16-batch x 16-point output tile:
//   out[b,p] = sum_l sum_{q in rect_l} bw_l(q,p) * ( z_l[b,:,q] . feat_l[:,p] )
// Inner dot over 64 channels = 16 chained V_WMMA_F32_16X16X4_F32.
__global__ __launch_bounds__(256)
void k_sample_wmma(const float* __restrict__ x, const float* __restrict__ grid,
                   const float* __restrict__ feat, const float* __restrict__ bias,
                   const float* __restrict__ z1, const float* __restrict__ z2,
                   const int* __restrict__ meta, float* __restrict__ out) {
    __shared__ float sfeat[64 * 128];   // feat tile for current level, all 128 pts of WG

    const int tid   = threadIdx.x;
    const int wave  = tid >> 5;
    const int lane  = tid & 31;
    const int col   = lane & 15;             // M row for A, N col for B/D
    const int kpair = (lane >> 4) << 1;      // K sub-pair: 0 (lanes 0-15) or 2 (16-31)
    const int pwg0  = blockIdx.x * 128;
    const int pcolWG = wave * 16 + col;
    const int p     = pwg0 + pcolWG;
    const int b0    = blockIdx.y * 16;
    const int bt    = b0 + col;              // this lane's A-matrix row (batch-time)

    float gx = fminf(fmaxf(grid[2 * p + 0], -1.f), 1.f);
    float gy = fminf(fmaxf(grid[2 * p + 1], -1.f), 1.f);

    v8f out_acc = {0.f, 0.f, 0.f, 0.f, 0.f, 0.f, 0.f, 0.f};

#if defined(__HIP_DEVICE_COMPILE__)
#pragma unroll
    for (int l = 0; l < 3; ++l) {
        const int W  = WH >> l;
        const int CS = (l == 0) ? (TT * WH * WH) : ((l == 1) ? 1024 : 256);  // channel stride
        const float* zbase = (l == 0) ? x : ((l == 1) ? z1 : z2);
        const int baseA = (l == 0)
            ? ((bt >> 4) * (CC * TT * WH * WH) + (bt & 15) * (WH * WH))
            : ((l == 1) ? bt * (CC * 1024) : bt * (CC * 256));
        const float* Abase = zbase + baseA;

        // Stage this level's feat block (64 rows x 128 points = 32 KB) into LDS.
        __syncthreads();
#if HAVE_ASYNC_LDS
        // Async global->LDS DMA path (ASYNCcnt): 8 x b128 per thread, no VGPR round-trip.
#pragma unroll
        for (int k = 0; k < 8; ++k) {
            int ch = tid + 256 * k;             // 2048 b128 chunks total
            int r  = ch >> 5;                   // feat row within level (0..63)
            int c4 = (ch & 31) << 2;            // point column (x4 floats)
            __builtin_amdgcn_global_load_async_to_lds_b128(
                GPTR(feat + (l * 64 + r) * PP + pwg0 + c4),
                LPTR(sfeat + r * 128 + c4),
                0, 0);
        }
        __builtin_amdgcn_s_wait_asynccnt(0);    // our async writes landed in LDS
#else
        for (int i = tid; i < 64 * 128; i += 256) {
            int r = i >> 7, ci = i & 127;
            sfeat[i] = feat[(l * 64 + r) * PP + pwg0 + ci];
        }
#endif
        __syncthreads();

        // Hoist the B operand (feat column for this lane's point) into registers.
        v2f Breg[16];
#pragma unroll
        for (int s = 0; s < 16; ++s) {
            Breg[s].x = sfeat[(4 * s + kpair) * 128 + pcolWG];
            Breg[s].y = sfeat[(4 * s + kpair + 1) * 128 + pcolWG];
        }

        // Per-lane bilinear corner data (for this lane's point).
        float fw  = (float)W;
        float ix  = ((gx + 1.f) * fw - 1.f) * 0.5f;
        float iy  = ((gy + 1.f) * fw - 1.f) * 0.5f;
        float ix0f = floorf(ix), iy0f = floorf(iy);
        float fx1 = ix - ix0f, fx0 = 1.f - fx1;
        float fy1 = iy - iy0f, fy0 = 1.f - fy1;
        int ix0 = (int)ix0f, iy0 = (int)iy0f;
        int ix1 = ix0 + 1, iy1 = iy0 + 1;
        float wx0 = (ix0 >= 0 && ix0 < W) ? fx0 : 0.f;
        float wx1 = (ix1 >= 0 && ix1 < W) ? fx1 : 0.f;
        float wy0 = (iy0 >= 0 && iy0 < W) ? fy0 : 0.f;
        float wy1 = (iy1 >= 0 && iy1 < W) ? fy1 : 0.f;
        int xc0 = min(max(ix0, 0), W - 1), xc1 = min(max(ix1, 0), W - 1);
        int yc0 = min(max(iy0, 0), W - 1), yc1 = min(max(iy1, 0), W - 1);

        const int xmin = meta[l * 4 + 0], xmax = meta[l * 4 + 1];
        const int ymin = meta[l * 4 + 2], ymax = meta[l * 4 + 3];

        for (int qy = ymin; qy <= ymax; ++qy) {
            float wqy = (qy == yc0 ? wy0 : 0.f) + (qy == yc1 ? wy1 : 0.f);
            for (int qx = xmin; qx <= xmax; ++qx) {
                float wqx = (qx == xc0 ? wx0 : 0.f) + (qx == xc1 ? wx1 : 0.f);
                float bw = wqx * wqy;
                if (!__any(bw != 0.f)) continue;   // wave-uniform skip, EXEC stays full
                const int q = qy * W + qx;

                v8f accp = {0.f, 0.f, 0.f, 0.f, 0.f, 0.f, 0.f, 0.f};
#pragma unroll
                for (int s = 0; s < 16; ++s) {
                    v2f a;
                    a.x = Abase[(4 * s + kpair) * CS + q];
                    a.y = Abase[(4 * s + kpair + 1) * CS + q];
                    accp = __builtin_amdgcn_wmma_f32_16x16x4_f32(
                        /*neg_a=*/false, a, /*neg_b=*/false, Breg[s],
                        /*c_mod=*/(short)0, accp, /*reuse_a=*/false, /*reuse_b=*/false);
                }
                // per-column bilinear weight (same for all 8 rows this lane holds)
#pragma unroll
                for (int r = 0; r < 8; ++r) out_acc[r] += bw * accp[r];
            }
        }
    }
#endif  // __HIP_DEVICE_COMPILE__

    // D layout: VGPR r -> row M=r (lanes 0-15) or M=r+8 (lanes 16-31), col N = lane%16.
    const float bv = bias[p];
    const int rowbase = (lane >> 4) << 3;
#pragma unroll
    for (int r = 0; r < 8; ++r) {
        out[(b0 + rowbase + r) * PP + p] = out_acc[r] + bv;
    }
}

// ---------------------------------------------------------------------------
extern "C" void kernel_launch(void* const* d_in, const int* in_sizes, int n_in,
                              void* d_out, int out_size, void* d_ws, size_t ws_size,
                              hipStream_t stream) {
    (void)in_sizes; (void)n_in; (void)out_size; (void)ws_size;
    const float* x    = (const float*)d_in[0];
    const float* grid = (const float*)d_in[1];
    const float* feat = (const float*)d_in[2];
    const float* bias = (const float*)d_in[3];
    float* out = (float*)d_out;

    float* z1 = (float*)((char*)d_ws + Z1_OFF);
    float* z2 = (float*)((char*)d_ws + Z2_OFF);
    int* meta = (int*)((char*)d_ws + META_OFF);

    k_init_meta<<<1, 32, 0, stream>>>(meta);
    k_bbox<<<(PP + 255) / 256, 256, 0, stream>>>(grid, meta);
    k_pool1<<<Z1_ELEMS / 256, 256, 0, stream>>>(x, z1);
    k_pool2<<<Z2_ELEMS / 256, 256, 0, stream>>>(z1, z2);

    dim3 g(16, 16);  // 16 point-groups x 16 batch-tiles
    k_sample_wmma<<<g, 256, 0, stream>>>(x, grid, feat, bias, z1, z2, meta, out);
}